// FlowCoding_35167192219966
// MI455X (gfx1250) — compile-verified
//
#include <hip/hip_runtime.h>
#include <hip/hip_fp16.h>

// FlowCoding for MI455X (gfx1250).
// Both bilinear resizes in the reference are identity (Hc,Wc == H,W), so the
// op reduces to: fp16 cast -> segment-sum + counts -> per-bucket fp16 mean ->
// gather. Pure bandwidth/atomic problem (~212 MB HBM traffic, ~10 us floor at
// 23.3 TB/s). The 16 MB accumulator (4 halves per bucket: c0,c1,c2,count)
// stays resident in the 192 MB L2, so the random scatter/gather never touches
// HBM. Scatter uses native packed fp16 atomics (global_atomic_pk_add_f16,
// forced via inline asm) = 2 atomics per element, matching the reference's
// fp16 segment_sum semantics. Index tiles are staged through the CDNA5 async
// global->LDS engine (ASYNCcnt path).

namespace {

constexpr int T_DIM  = 16;
constexpr int C_DIM  = 3;
constexpr int H_DIM  = 480;
constexpr int W_DIM  = 864;
constexpr int HW     = H_DIM * W_DIM;        // 414720
constexpr int N_ELEM = T_DIM * HW;           // 6635520 = 25920 * 256
constexpr int NPIX   = 2000000;
constexpr int BLOCK  = 256;
constexpr int BPT    = HW / BLOCK;           // 1620 blocks per t-slice (exact)

__device__ __forceinline__ int async_load_idx(const int* __restrict__ enc,
                                              int i, int* sidx, int tid)
{
    // CDNA5 async global->LDS copy of this block's 256 indices (ASYNCcnt).
    unsigned long long gaddr =
        (unsigned long long)(uintptr_t)(const void*)(enc + i);
    unsigned int laddr =
        (unsigned int)(unsigned long long)(uintptr_t)(void*)(&sidx[tid]);
    asm volatile("global_load_async_to_lds_b32 %0, %1, off"
                 :: "v"(laddr), "v"(gaddr) : "memory");
    asm volatile("s_wait_asynccnt 0" ::: "memory");
    return sidx[tid];
}

// ---------------------------------------------------------------------------
// Pass A: scatter-accumulate. One thread per (t,h,w) element.
// acc[idx] record layout (8B): half2{c0,c1}, half2{c2,count}
// ---------------------------------------------------------------------------
__global__ __launch_bounds__(BLOCK) void fc_scatter(
    const float* __restrict__ samples,
    const int*   __restrict__ enc,
    __half2*     __restrict__ acc)
{
    __shared__ int sidx[BLOCK];
    const int tid = threadIdx.x;
    const int b   = blockIdx.x;
    const int i   = b * BLOCK + tid;            // exact grid, no tail

    const int idx = async_load_idx(enc, i, sidx, tid);

    // t / hw from the uniform block index -> SALU, no per-lane division.
    const int t  = b / BPT;
    const int hw = (b - t * BPT) * BLOCK + tid;

    const float* sp = samples + (size_t)t * (size_t)(C_DIM * HW) + hw;
    const float f0 = sp[0];
    const float f1 = sp[HW];
    const float f2 = sp[2 * HW];

    __half2 ab = __halves2half2(__float2half(f0), __float2half(f1));
    __half2 c1 = __halves2half2(__float2half(f2), __float2half(1.0f));

    const unsigned long long a =
        (unsigned long long)(uintptr_t)(void*)(acc + 2 * (size_t)idx);
    const unsigned int d0 = *reinterpret_cast<unsigned int*>(&ab);
    const unsigned int d1 = *reinterpret_cast<unsigned int*>(&c1);

    // Native packed-fp16 atomic adds (no-return form, STOREcnt-tracked).
    asm volatile("global_atomic_pk_add_f16 %0, %1, off"
                 :: "v"(a), "v"(d0) : "memory");
    asm volatile("global_atomic_pk_add_f16 %0, %1, off offset:4"
                 :: "v"(a), "v"(d1) : "memory");
}

// ---------------------------------------------------------------------------
// Pass B: convert sums -> means, once per bucket (in fp16, like reference).
// ---------------------------------------------------------------------------
__global__ __launch_bounds__(BLOCK) void fc_mean(float2* __restrict__ acc)
{
    const int p = blockIdx.x * BLOCK + threadIdx.x;
    if (p >= NPIX) return;
    float2 r = acc[p];
    __half2 ab = *reinterpret_cast<__half2*>(&r.x);
    __half2 cd = *reinterpret_cast<__half2*>(&r.y);
    const __half cnt = __high2half(cd);
    if (__half2float(cnt) > 0.0f) {
        const __half2 cc = __halves2half2(cnt, cnt);
        ab = __h2div(ab, cc);
        const __half m2 = __hdiv(__low2half(cd), cnt);
        cd = __halves2half2(m2, cnt);
        r.x = *reinterpret_cast<float*>(&ab);
        r.y = *reinterpret_cast<float*>(&cd);
        acc[p] = r;
    }
}

// ---------------------------------------------------------------------------
// Pass C: gather means back through the index map; write (T,C,H,W) floats.
// Random 8B reads hit L2 (16 MB working set); output stores are coalesced.
// ---------------------------------------------------------------------------
__global__ __launch_bounds__(BLOCK) void fc_gather(
    const int*    __restrict__ enc,
    const float2* __restrict__ acc,
    float*        __restrict__ out)
{
    __shared__ int sidx[BLOCK];
    const int tid = threadIdx.x;
    const int b   = blockIdx.x;
    const int i   = b * BLOCK + tid;            // exact grid

    const int idx = async_load_idx(enc, i, sidx, tid);

    float2 r = acc[idx];
    const __half2 ab = *reinterpret_cast<const __half2*>(&r.x);
    const __half2 cd = *reinterpret_cast<const __half2*>(&r.y);

    const int t  = b / BPT;
    const int hw = (b - t * BPT) * BLOCK + tid;
    float* op = out + (size_t)t * (size_t)(C_DIM * HW) + hw;
    op[0]      = __half2float(__low2half(ab));
    op[HW]     = __half2float(__high2half(ab));
    op[2 * HW] = __half2float(__low2half(cd));
}

} // namespace

extern "C" void kernel_launch(void* const* d_in, const int* in_sizes, int n_in,
                              void* d_out, int out_size, void* d_ws, size_t ws_size,
                              hipStream_t stream)
{
    (void)in_sizes; (void)n_in; (void)out_size; (void)ws_size;

    const float* samples = (const float*)d_in[0];   // (16,3,480,864) f32
    const int*   enc     = (const int*)d_in[1];     // (1,16,480,864) i32
    // d_in[2] = n_pixels scalar (constant 2,000,000)

    __half2* acc = (__half2*)d_ws;                  // 2M * 8B = 16 MB scratch

    hipMemsetAsync(d_ws, 0, (size_t)NPIX * 8, stream);

    fc_scatter<<<N_ELEM / BLOCK, BLOCK, 0, stream>>>(samples, enc, acc);
    fc_mean   <<<(NPIX + BLOCK - 1) / BLOCK, BLOCK, 0, stream>>>((float2*)d_ws);
    fc_gather <<<N_ELEM / BLOCK, BLOCK, 0, stream>>>(enc, (const float2*)d_ws,
                                                     (float*)d_out);
}